// RipsModel_63642825392501
// MI455X (gfx1250) — compile-verified
//
#include <hip/hip_runtime.h>

// CDNA5 / gfx1250: wave32, WMMA f32 16x16x4.
typedef float v2f __attribute__((ext_vector_type(2)));
typedef float v8f __attribute__((ext_vector_type(8)));

#define DIM 512  // feature dimension d from the reference (n=4096, d=512)

// One WMMA accumulation step: load 2 consecutive K-elements of both rows,
// form the difference (a slice of the 16x4 diff A-tile), and do C += A*A^T.
// The same 2-VGPR operand is legal as both A and B because B's layout
// (lane=N, VGPR=K) is the transpose-mapping of A's (lane=M, VGPR=K).
__device__ __forceinline__ v8f wmma_step(const float* __restrict__ a,
                                         const float* __restrict__ b,
                                         int k, v8f c) {
  v2f av = *(const v2f*)(a + k);   // global_load_b64, 8B aligned (k is even)
  v2f bv = *(const v2f*)(b + k);
  v2f d  = av - bv;
  return __builtin_amdgcn_wmma_f32_16x16x4_f32(
      /*neg_a=*/false, d, /*neg_b=*/false, d,
      /*c_mod=*/(short)0, c, /*reuse_a=*/false, /*reuse_b=*/false);
}

// One wave (32 threads) per tile of 16 persistence pairs.
// Lane L and lane L+16 cooperatively own pair m = L&15:
//   lanes 0-15 supply K-slice {kc, kc+1}, lanes 16-31 supply {kc+2, kc+3}
// per the ISA's 32-bit 16x4 A-matrix VGPR layout.
__global__ __launch_bounds__(32)
void rips_pair_dist_wmma(const float* __restrict__ X,
                         const int* __restrict__ ids,
                         float* __restrict__ out,
                         int npairs) {
  const int lane = threadIdx.x;        // 0..31 (wave32)
  const int tile = blockIdx.x;
  const int m    = lane & 15;          // pair-in-tile this lane serves
  const int p    = tile * 16 + m;      // global pair index
  const int pl   = (p < npairs) ? p : (npairs - 1);  // clamp padding lanes

  // Gather FIRST (the whole point vs. the reference): only 2 rows per pair.
  const int r0 = ids[2 * pl];
  const int r1 = ids[2 * pl + 1];
  const float* __restrict__ a = X + (size_t)r0 * DIM;
  const float* __restrict__ b = X + (size_t)r1 * DIM;

  const int koff = (lane < 16) ? 0 : 2;  // which K half-slice this lane owns

  // 4 independent accumulators to break the WMMA->WMMA RAW hazard chain.
  v8f c0 = {}, c1 = {}, c2 = {}, c3 = {};
  #pragma unroll 2
  for (int kc = 0; kc < DIM; kc += 16) {
    c0 = wmma_step(a, b, kc +  0 + koff, c0);
    c1 = wmma_step(a, b, kc +  4 + koff, c1);
    c2 = wmma_step(a, b, kc +  8 + koff, c2);
    c3 = wmma_step(a, b, kc + 12 + koff, c3);
  }
  v8f c = (c0 + c1) + (c2 + c3);   // C[m][n] = sum_k diff[m][k]*diff[n][k]

  // Diagonal extraction per the 16x16 f32 C/D layout:
  //   lanes 0-15:  VGPR v holds M=v,   N=lane      -> diag at lanes 0-7, v=lane
  //   lanes 16-31: VGPR v holds M=v+8, N=lane-16   -> diag at lanes 24-31, v=lane-24
  const bool lo      = (lane < 16);
  const bool hasDiag = lo ? (lane < 8) : (lane >= 24);
  const int  v       = lo ? lane : (lane - 24);
  float d2 = c[0];
  #pragma unroll
  for (int i = 1; i < 8; ++i) d2 = (v == i) ? c[i] : d2;

  // dgm flat element p = DX[ids[2p], ids[2p+1]] = ||x_r0 - x_r1||
  if (hasDiag && p < npairs) out[p] = sqrtf(fmaxf(d2, 0.0f));
}

extern "C" void kernel_launch(void* const* d_in, const int* in_sizes, int n_in,
                              void* d_out, int out_size, void* d_ws, size_t ws_size,
                              hipStream_t stream) {
  const float* X   = (const float*)d_in[0];   // [4096, 512] f32
  const int*   ids = (const int*)d_in[1];     // [4*CARD] i32
  float*       out = (float*)d_out;           // [CARD, 2] f32 flat = npairs scalars

  const int npairs = in_sizes[1] / 2;         // 4*CARD ids -> 2*CARD (row,col) pairs
  const int tiles  = (npairs + 15) / 16;      // 16 pairs per wave
  rips_pair_dist_wmma<<<tiles, 32, 0, stream>>>(X, ids, out, npairs);
}